// PositionalEncoding2D_24489903522576
// MI455X (gfx1250) — compile-verified
//
#include <hip/hip_runtime.h>

// out[b,s,d] = x[b,s,d] + pe[s,d]
// x: [8, 4096, 1024] f32.  Memory-bound: 256 MiB total traffic -> ~11.5us @ 23.3 TB/s.
// One block per row s; pe row-chunk computed once in registers (branchless,
// cos(x) = sin(x + pi/2)); the 8 batch rows stream through LDS via gfx1250
// async global->LDS B128 loads (ASYNCcnt), double-buffered, NT hints both ways.

#define BATCH 8
#define SEQ   4096
#define DIM   1024
#define HALF  512
// log2(10000)
#define L2_10K  13.287712379549449f
#define HALF_PI 1.57079632679489662f

typedef float v4f __attribute__((ext_vector_type(4)));

__device__ __forceinline__ float pe_val(int s, int d) {
    // d <  HALF: sin(s * 10000^-(2d/DIM))
    // d >= HALF: cos(s * 10000^-((2(d-HALF)+1)/DIM)) = sin(arg + pi/2)
    const bool hi   = (d >= HALF);
    const float num = (float)(2 * d - (hi ? (2 * HALF - 1) : 0));
    const float f   = __builtin_amdgcn_exp2f(-(num * (1.0f / (float)DIM)) * L2_10K); // v_exp_f32
    const float ph  = hi ? HALF_PI : 0.0f;
    return __sinf((float)s * f + ph);                                                // v_sin_f32
}

__device__ __forceinline__ void async_load_b128_nt(unsigned lds_byte_off,
                                                   unsigned gbl_byte_off,
                                                   const float* base) {
    // GVS mode: mem_addr = SGPR_u64(base) + VGPR_i32(gbl_byte_off); LDS dest = VGPR(lds_byte_off)
    // Streaming data (read exactly once) -> non-temporal hint.
    asm volatile("global_load_async_to_lds_b128 %0, %1, %2 offset:0 th:TH_LOAD_NT"
                 :
                 : "v"(lds_byte_off), "v"(gbl_byte_off), "s"(base)
                 : "memory");
}

__global__ void __launch_bounds__(256)
pe_add_kernel(const float* __restrict__ x, float* __restrict__ out) {
    __shared__ float buf[2][DIM];   // 8 KB, double buffer (4 KB per batch row)

    const int s   = blockIdx.x;     // 0..4095
    const int tid = threadIdx.x;    // 0..255
    const int d0  = tid * 4;        // 4 consecutive floats per thread

    // ---- positional-encoding values for this (s, d0..d0+3), computed once ----
    float pe[4];
#pragma unroll
    for (int k = 0; k < 4; ++k) pe[k] = pe_val(s, d0 + k);

    // ---- LDS byte offsets (generic ptr low 32 bits == wave-relative LDS offset) ----
    const unsigned lds_base = (unsigned)(uintptr_t)(&buf[0][0]);
    const unsigned lds_off0 = lds_base + (unsigned)(tid * 16);
    const unsigned lds_off1 = lds_base + 4096u + (unsigned)(tid * 16);

    // Global byte offset of this thread's chunk in batch 0; per-batch stride = S*D*4.
    const unsigned g0      = (unsigned)((s * DIM + d0) * 4);
    const unsigned bstride = (unsigned)(SEQ * DIM * 4);   // 16 MiB

    // ---- prime the pipeline: batch 0 -> buf[0] ----
    async_load_b128_nt(lds_off0, g0, x);

#pragma unroll
    for (int b = 0; b < BATCH; ++b) {
        // issue next batch into the other buffer while we consume this one
        if (b + 1 < BATCH) {
            async_load_b128_nt((b & 1) ? lds_off0 : lds_off1,
                               g0 + (unsigned)(b + 1) * bstride, x);
            asm volatile("s_wait_asynccnt 1" ::: "memory"); // batch b landed (in-order)
        } else {
            asm volatile("s_wait_asynccnt 0" ::: "memory");
        }
        __syncthreads();  // every wave's chunk of buf[b&1] is visible

        const v4f v = *reinterpret_cast<const v4f*>(&buf[b & 1][d0]);
        v4f r;
        r.x = v.x + pe[0];
        r.y = v.y + pe[1];
        r.z = v.z + pe[2];
        r.w = v.w + pe[3];

        // Written exactly once, never re-read -> non-temporal store (keep L2 clean).
        __builtin_nontemporal_store(
            r, reinterpret_cast<v4f*>(
                   &out[(size_t)b * (SEQ * DIM) + (size_t)s * DIM + d0]));

        __syncthreads();  // don't let the next async overwrite a buffer still being read
    }
}

extern "C" void kernel_launch(void* const* d_in, const int* in_sizes, int n_in,
                              void* d_out, int out_size, void* d_ws, size_t ws_size,
                              hipStream_t stream) {
    (void)in_sizes; (void)n_in; (void)d_ws; (void)ws_size; (void)out_size;
    const float* x = (const float*)d_in[0];
    float* out     = (float*)d_out;
    pe_add_kernel<<<SEQ, 256, 0, stream>>>(x, out);
}